// Loss_67637144978489
// MI455X (gfx1250) — compile-verified
//
#include <hip/hip_runtime.h>
#include <hip/hip_bf16.h>
#include <math.h>

typedef __attribute__((ext_vector_type(2))) float v2f;
typedef __attribute__((ext_vector_type(8))) float v8f;

#define BB 16
#define QQ 100
#define GG 100
#define CC 100
#define EPSK 1e-7f

// iou and background-ratio between two boxes in (cx, cy, w, h) form
__device__ __forceinline__ void box_terms(const float bp[4], const float bg[4],
                                          float& iou, float& bgr) {
    float pulx = bp[0] - 0.5f * bp[2], puly = bp[1] - 0.5f * bp[3];
    float pdrx = bp[0] + 0.5f * bp[2], pdry = bp[1] + 0.5f * bp[3];
    float gulx = bg[0] - 0.5f * bg[2], guly = bg[1] - 0.5f * bg[3];
    float gdrx = bg[0] + 0.5f * bg[2], gdry = bg[1] + 0.5f * bg[3];

    float iw = fmaxf(fminf(pdrx, gdrx) - fmaxf(pulx, gulx) + 1.0f, 0.0f);
    float ih = fmaxf(fminf(pdry, gdry) - fmaxf(puly, guly) + 1.0f, 0.0f);
    float inter = iw * ih;

    float pw = fmaxf(pdrx - pulx + 1.0f, 0.0f);
    float ph = fmaxf(pdry - puly + 1.0f, 0.0f);
    float gw = fmaxf(gdrx - gulx + 1.0f, 0.0f);
    float gh = fmaxf(gdry - guly + 1.0f, 0.0f);
    float pa = pw * ph, ga = gw * gh;
    float uni = pa + ga - inter;
    iou = inter / fmaxf(uni, 1e-9f);

    float bw = fmaxf(fmaxf(pdrx, gdrx) - fminf(pulx, gulx) + 1.0f, 0.0f);
    float bh = fmaxf(fmaxf(pdry, gdry) - fminf(puly, guly) + 1.0f, 0.0f);
    float bound = bw * bh;
    bgr = (bound - uni) / fmaxf(bound, 1e-9f);
}

__global__ void zero_out_kernel(float* out) { out[0] = 0.0f; }

// ---------------------------------------------------------------------------
// Cost matrix: one wave per 16x16 (q,g) tile of one batch.
// probs tile computed as GEMM  labels_pred(16xC) x onehot(labels_gt)(Cx16)
// via V_WMMA_F32_16X16X4_F32, K stepped by 4 (C=100 -> 25 WMMAs, fully
// unrolled so the scheduler can clause the A-loads ahead of the WMMAs).
// ---------------------------------------------------------------------------
__global__ __launch_bounds__(32) void cost_kernel(
    const float* __restrict__ bbox_pred,    // (B,Q,4)
    const float* __restrict__ labels_pred,  // (B,Q,C)
    const float* __restrict__ bbox_gt,      // (B,G,4)
    const int*   __restrict__ labels_gt,    // (B,G)
    float* __restrict__ cost)               // (B,Q,G)
{
    const int b    = blockIdx.z;
    const int q0   = blockIdx.y * 16;
    const int g0   = blockIdx.x * 16;
    const int lane = threadIdx.x;      // 0..31, wave32
    const int half = lane >> 4;        // 0 or 1
    const int sub  = lane & 15;

    // A-fragment source row (clamped for the q>=100 fringe; results masked on store)
    int qa = q0 + sub; if (qa >= QQ) qa = QQ - 1;
    const float* Arow = labels_pred + ((size_t)b * QQ + qa) * CC;

    // B-fragment: one-hot of this lane's gt label (clamped fringe)
    int gb = g0 + sub; if (gb >= GG) gb = GG - 1;
    const int gl = labels_gt[b * GG + gb];

    v8f acc = {};
    #pragma unroll
    for (int k0 = 0; k0 < CC; k0 += 4) {
        // A 16x4 f32 layout: lanes 0-15 hold K=0,1 ; lanes 16-31 hold K=2,3
        v2f a;
        a.x = Arow[k0 + 2 * half + 0];
        a.y = Arow[k0 + 2 * half + 1];
        // B 4x16 f32 layout: lanes 0-15 hold rows K=0,1 of column n ; 16-31 rows K=2,3
        v2f bm;
        bm.x = (gl == (k0 + 2 * half + 0)) ? 1.0f : 0.0f;
        bm.y = (gl == (k0 + 2 * half + 1)) ? 1.0f : 0.0f;
        acc = __builtin_amdgcn_wmma_f32_16x16x4_f32(
            /*neg_a=*/false, a, /*neg_b=*/false, bm,
            /*c_mod=*/(short)0, acc, /*reuse_a=*/false, /*reuse_b=*/false);
    }

    // D layout: VGPR r holds M = r (lanes 0-15) / r+8 (lanes 16-31), N = sub
    const int g = g0 + sub;
    float bg[4] = {0.f, 0.f, 0.f, 0.f};
    if (g < GG) {
        const float* p = bbox_gt + ((size_t)b * GG + g) * 4;
        bg[0] = p[0]; bg[1] = p[1]; bg[2] = p[2]; bg[3] = p[3];
    }
    #pragma unroll
    for (int r = 0; r < 8; ++r) {
        int q = q0 + r + 8 * half;
        if (q < QQ && g < GG) {
            const float* pp = bbox_pred + ((size_t)b * QQ + q) * 4;
            float bp[4] = {pp[0], pp[1], pp[2], pp[3]};
            float l1 = fabsf(bp[0] - bg[0]) + fabsf(bp[1] - bg[1]) +
                       fabsf(bp[2] - bg[2]) + fabsf(bp[3] - bg[3]);
            float iou, bgr;
            box_terms(bp, bg, iou, bgr);
            cost[((size_t)b * QQ + q) * GG + g] = l1 - (iou - bgr) - acc[r];
        }
    }
}

// ---------------------------------------------------------------------------
// Hungarian (Jonker-Volgenant) per batch. ONE WAVE per batch: in a
// single-wave workgroup the ISA lowers barrier ops to S_NOP, so every
// __syncthreads below costs only an LDS-counter wait — no inter-wave
// barrier round trips in the sequential inner loop.
// ---------------------------------------------------------------------------
#define HN  QQ
#define HN1 (QQ + 1)
#define HT  32

__global__ __launch_bounds__(HT) void hungarian_kernel(
    const float* __restrict__ cost,   // (B, n, n)
    int* __restrict__ col)            // (B, n)  col[row] = matched column
{
    const int b = blockIdx.x;
    const int t = threadIdx.x;
    const float* C = cost + (size_t)b * HN * HN;

    __shared__ float u[HN1], v[HN1], minv[HN1];
    __shared__ int   p[HN1], way[HN1], used[HN1];
    __shared__ int   sj0;
    __shared__ float rmin[HT];
    __shared__ int   rarg[HT];

    for (int j = t; j <= HN; j += HT) { u[j] = 0.f; v[j] = 0.f; p[j] = 0; way[j] = 0; }
    __syncthreads();

    for (int i = 1; i <= HN; ++i) {
        if (t == 0) { p[0] = i; sj0 = 0; }
        for (int j = t; j <= HN; j += HT) { minv[j] = 1e30f; used[j] = 0; }
        __syncthreads();

        for (;;) {
            const int j0 = sj0;
            if (t == 0) used[j0] = 1;
            __syncthreads();
            const int   i0  = p[j0];
            const float ui0 = u[i0];
            const float* Crow = C + (size_t)(i0 - 1) * HN;

            // relax
            for (int j = 1 + t; j <= HN; j += HT) {
                if (!used[j]) {
                    float cur = Crow[j - 1] - ui0 - v[j];
                    if (cur < minv[j]) { minv[j] = cur; way[j] = j0; }
                }
            }
            __syncthreads();

            // argmin over unused columns
            float best = 1e30f; int bestj = 1;
            for (int j = 1 + t; j <= HN; j += HT)
                if (!used[j] && minv[j] < best) { best = minv[j]; bestj = j; }
            rmin[t] = best; rarg[t] = bestj;
            __syncthreads();
            for (int s = HT / 2; s > 0; s >>= 1) {
                if (t < s && rmin[t + s] < rmin[t]) { rmin[t] = rmin[t + s]; rarg[t] = rarg[t + s]; }
                __syncthreads();
            }
            const float delta = rmin[0];
            const int   j1    = rarg[0];

            // update potentials (p[j] distinct over used j -> no write conflicts)
            for (int j = t; j <= HN; j += HT) {
                if (used[j]) { u[p[j]] += delta; v[j] -= delta; }
                else         { minv[j] -= delta; }
            }
            __syncthreads();
            if (t == 0) sj0 = j1;
            __syncthreads();
            if (p[j1] == 0) break;
        }

        // augment along alternating path (cheap, sequential)
        if (t == 0) {
            int j0 = sj0;
            while (j0 != 0) { int j1 = way[j0]; p[j0] = p[j1]; j0 = j1; }
        }
        __syncthreads();
    }

    for (int j = 1 + t; j <= HN; j += HT)
        col[b * HN + (p[j] - 1)] = j - 1;
}

// ---------------------------------------------------------------------------
// Losses: one block per batch; thread q handles query row q.
// ---------------------------------------------------------------------------
__global__ __launch_bounds__(128) void loss_kernel(
    const float* __restrict__ bbox_pred,
    const float* __restrict__ labels_pred,
    const float* __restrict__ bbox_gt,
    const int*   __restrict__ labels_gt,
    const int*   __restrict__ col,
    float* __restrict__ out)
{
    const int b = blockIdx.x;
    const int t = threadIdx.x;

    float nll = 0.f, reg = 0.f, giou = 0.f;
    if (t < QQ) {
        const int q  = t;
        const int c  = col[b * QQ + q];
        const int gl = labels_gt[b * GG + c];
        const float* row = labels_pred + ((size_t)b * QQ + q) * CC;

        // log-softmax of log(clip(p)) at gl
        float mx = -1e30f;
        for (int j = 0; j < CC; ++j) {
            float lg = __logf(fminf(fmaxf(row[j], EPSK), 1.0f - EPSK));
            mx = fmaxf(mx, lg);
        }
        float s = 0.f, tgt = 0.f;
        for (int j = 0; j < CC; ++j) {
            float lg = __logf(fminf(fmaxf(row[j], EPSK), 1.0f - EPSK));
            s += __expf(lg - mx);
            if (j == gl) tgt = lg;
        }
        nll = -(tgt - (mx + __logf(s)));

        // matched L1
        const float* pp = bbox_pred + ((size_t)b * QQ + q) * 4;
        const float* tg = bbox_gt   + ((size_t)b * GG + c) * 4;
        reg = fabsf(pp[0] - tg[0]) + fabsf(pp[1] - tg[1]) +
              fabsf(pp[2] - tg[2]) + fabsf(pp[3] - tg[3]);

        // diagonal giou term (reference pairs pred[q] with gt[q])
        const float* dg = bbox_gt + ((size_t)b * GG + q) * 4;
        float bp[4] = {pp[0], pp[1], pp[2], pp[3]};
        float bg[4] = {dg[0], dg[1], dg[2], dg[3]};
        float iou, bgr;
        box_terms(bp, bg, iou, bgr);
        giou = iou - bgr;
    }

    __shared__ float r0[128], r1[128], r2[128];
    r0[t] = nll; r1[t] = reg; r2[t] = giou;
    __syncthreads();
    for (int s = 64; s > 0; s >>= 1) {
        if (t < s) { r0[t] += r0[t + s]; r1[t] += r1[t + s]; r2[t] += r2[t + s]; }
        __syncthreads();
    }
    if (t == 0) {
        float label_loss = r0[0] / (float)QQ;
        float reg_loss   = r1[0] / (float)(QQ * 4);
        float giou_mean  = r2[0] / (float)QQ;
        atomicAdd(out, 1.0f * label_loss + 5.0f * reg_loss + 2.0f * giou_mean);
    }
}

extern "C" void kernel_launch(void* const* d_in, const int* in_sizes, int n_in,
                              void* d_out, int out_size, void* d_ws, size_t ws_size,
                              hipStream_t stream) {
    const float* bbox_pred   = (const float*)d_in[0];  // (16,100,4)
    const float* labels_pred = (const float*)d_in[1];  // (16,100,100)
    const float* bbox_gt     = (const float*)d_in[2];  // (16,100,4)
    const int*   labels_gt   = (const int*)  d_in[3];  // (16,100)
    float* out = (float*)d_out;

    float* cost = (float*)d_ws;                                        // 640000 B
    int*   col  = (int*)((char*)d_ws + (size_t)BB * QQ * GG * sizeof(float)); // 6400 B

    zero_out_kernel<<<1, 1, 0, stream>>>(out);

    dim3 gcost((GG + 15) / 16, (QQ + 15) / 16, BB);   // 7 x 7 x 16 waves
    cost_kernel<<<gcost, 32, 0, stream>>>(bbox_pred, labels_pred, bbox_gt,
                                          labels_gt, cost);

    hungarian_kernel<<<BB, HT, 0, stream>>>(cost, col);

    loss_kernel<<<BB, 128, 0, stream>>>(bbox_pred, labels_pred, bbox_gt,
                                        labels_gt, col, out);
}